// CustomInverseSTFT_90864328114847
// MI455X (gfx1250) — compile-verified
//
#include <hip/hip_runtime.h>
#include <stdint.h>

// Problem constants (match reference)
#define NFRAMES   8192
#define FLEN      1024       // FRAME_LENGTH == FFT_LENGTH
#define STEP      256
#define NTILES    8195       // out_len / STEP ; out_len = 256*8191 + 1024 = 2097920
#define THETA0    6.1359231515425649e-03f   // 2*pi/1024
#define INV_N     9.765625e-04f             // 1/1024

#define NBLOCKS   2048
#define NTHREADS  256

// Per-lane 16B async copy global -> LDS (CDNA5 ASYNCcnt path, CDNA5 ISA).
__device__ __forceinline__ void async_g2l_b128(unsigned lds_byte_addr, const float* gptr) {
    asm volatile("global_load_async_to_lds_b128 %0, %1, off"
                 :
                 : "v"(lds_byte_addr), "v"((unsigned long long)(uintptr_t)gptr)
                 : "memory");
}

__device__ __forceinline__ void wait_asynccnt0() {
    asm volatile("s_wait_asynccnt 0x0" ::: "memory");
}

__global__ __launch_bounds__(NTHREADS)
void CustomInverseSTFT_90864328114847_kernel(const float* __restrict__ Sr,
                                             const float* __restrict__ Si,
                                             float* __restrict__ out) {
    // [buf][seg = 2*d + {0:real,1:imag}][elem]  -> 2 * 8 * 256 * 4B = 16 KB
    __shared__ float sbuf[2][8][NTHREADS];

    const int tid = threadIdx.x;
    // Flat address of an LDS object: low 32 bits are the wave-relative LDS byte
    // offset -- what async-to-LDS wants.
    const unsigned lds_base = (unsigned)(uintptr_t)&sbuf[0][0][0];

    // Fill one 8 KB tile buffer with 2 x b128 async copies per thread.
    // Chunk c in [0,512): 16B each, LDS-linear at c*16.
    //   d   = c >> 7          (overlap index 0..3; wave-uniform)
    //   arr = (c >> 6) & 1    (0 = real, 1 = imag)
    //   elem within segment = (c & 63) * 4
    auto load_tile = [&](int tt, int b) {
        #pragma unroll
        for (int h = 0; h < 2; ++h) {
            const int c = tid + (h << 8);
            const int d = c >> 7;
            const int i = tt - d;                       // contributing frame
            if (i >= 0 && i < NFRAMES) {
                const int arr = (c >> 6) & 1;
                const int e   = (c & 63) << 2;          // element offset in segment
                const float* g = (arr ? Si : Sr) + ((size_t)i << 10) + (d << 8) + e;
                const unsigned l = lds_base + ((unsigned)b << 13) + ((unsigned)c << 4);
                async_g2l_b128(l, g);
            }
        }
    };

    int tile = (int)blockIdx.x;
    int b    = 0;
    load_tile(tile, 0);                                 // prologue: prime the pipeline

    for (; tile < NTILES; tile += (int)gridDim.x) {
        wait_asynccnt0();                               // this wave's copies for `tile` done
        __syncthreads();                                // all waves' copies done; prev compute done

        const int nxt = tile + (int)gridDim.x;          // prefetch next tile into other buffer
        if (nxt < NTILES) load_tile(nxt, b ^ 1);

        // One sincos per thread; the 4 per-d angles differ by pi/2:
        //   theta_d = theta0 + d*pi/2  =>  quarter-turn sign/swap identities.
        float s0, c0;
        __sincosf((float)tid * THETA0, &s0, &c0);
        const float cd[4] = {  c0, -s0, -c0,  s0 };
        const float sd[4] = {  s0,  c0, -s0, -c0 };

        float acc = 0.0f, wacc = 0.0f;
        #pragma unroll
        for (int d = 0; d < 4; ++d) {
            const int i = tile - d;
            if (i >= 0 && i < NFRAMES) {
                const float r  = sbuf[b][2 * d    ][tid];
                const float im = sbuf[b][2 * d + 1][tid];
                const float c = cd[d], s = sd[d];
                const float w  = fmaf(-0.5f, c, 0.5f);           // Hann: 0.5 - 0.5*cos
                const float tf = ((r - im) * c - (r + im) * s) * INV_N;
                acc  = fmaf(tf, w, acc);
                wacc += w;
            }
        }

        out[((size_t)tile << 8) + (size_t)tid] = __fdividef(acc, fmaxf(wacc, 1e-8f));
        b ^= 1;
    }
}

extern "C" void kernel_launch(void* const* d_in, const int* in_sizes, int n_in,
                              void* d_out, int out_size, void* d_ws, size_t ws_size,
                              hipStream_t stream) {
    (void)in_sizes; (void)n_in; (void)d_ws; (void)ws_size; (void)out_size;
    const float* Sr = (const float*)d_in[0];
    const float* Si = (const float*)d_in[1];
    float* out = (float*)d_out;
    CustomInverseSTFT_90864328114847_kernel<<<NBLOCKS, NTHREADS, 0, stream>>>(Sr, Si, out);
}